// Top_K_2525440770780
// MI455X (gfx1250) — compile-verified
//
#include <hip/hip_runtime.h>

// Top-K row mask (B=8192, N=8192, K=64), memory-bound: 512 MB traffic,
// ~22us floor @ 23.3 TB/s. One workgroup per row; row lives in LDS.
// CDNA5 paths used: GLOBAL_LOAD_ASYNC_TO_LDS_B128 / GLOBAL_STORE_ASYNC_FROM_LDS_B128
// (ASYNCcnt), s_wait_asynccnt, LDS ds_add_u32 atomics for radix select.

#define NCOLS 8192
#define TPB 256
#define WAVES 8
#define HSTRIDE 257   // pad per-wave histograms to decorrelate LDS banks

typedef float float4v __attribute__((ext_vector_type(4)));
typedef int   int4v   __attribute__((ext_vector_type(4)));
// Builtin parameter types (from hipcc diagnostic): int4 pointers w/ addrspace.
typedef __attribute__((address_space(1))) int4v* g_i4p;
typedef __attribute__((address_space(3))) int4v* l_i4p;

#if defined(__has_builtin)
#  if __has_builtin(__builtin_amdgcn_global_load_async_to_lds_b128) && \
      __has_builtin(__builtin_amdgcn_global_store_async_from_lds_b128)
#    define HAVE_ASYNC_LDS 1
#  endif
#  if __has_builtin(__builtin_amdgcn_s_wait_asynccnt)
#    define HAVE_WAIT_ASYNC 1
#  endif
#endif
#ifndef HAVE_ASYNC_LDS
#  define HAVE_ASYNC_LDS 0
#endif
#ifndef HAVE_WAIT_ASYNC
#  define HAVE_WAIT_ASYNC 0
#endif

__device__ __forceinline__ void wait_async0() {
#if HAVE_ASYNC_LDS
#  if HAVE_WAIT_ASYNC
  __builtin_amdgcn_s_wait_asynccnt(0);
#  else
  asm volatile("s_wait_asynccnt 0" ::: "memory");
#  endif
#endif
}

// XOR-swizzle at float4 (16B) granularity: keeps b128 transfers contiguous,
// makes both the strided (idx = j*256+t) and chunked (idx = t*32+j) LDS
// access patterns bank-conflict-(near-)free across a wave32.
__device__ __forceinline__ unsigned swzg(unsigned g) { return g ^ ((g >> 4) & 15u); }
__device__ __forceinline__ unsigned swzw(unsigned idx) { return (swzg(idx >> 2) << 2) | (idx & 3u); }

// Monotonic unsigned key for ReLU'd fp32: positive floats order as uints; x<=0 -> 0.
__device__ __forceinline__ unsigned key_of(float x) { return x > 0.0f ? __float_as_uint(x) : 0u; }

__global__ __launch_bounds__(TPB) void topk_mask_kernel(const float* __restrict__ x,
                                                        const int* __restrict__ kptr,
                                                        float* __restrict__ out) {
  __shared__ float rowbuf[NCOLS];             // 32 KB, swizzled layout
  __shared__ unsigned whist[WAVES * HSTRIDE]; // per-wave privatized histograms
  __shared__ unsigned hist[TPB];              // reduced hist / suffix-scan buffer
  __shared__ unsigned eqcnt[TPB];             // per-chunk equal counts
  __shared__ unsigned sel;                    // selected digit per radix pass

  const unsigned tid = threadIdx.x;
  const unsigned wv = tid >> 5;
  const unsigned row = blockIdx.x;
  const float* __restrict__ xrow = x + (size_t)row * NCOLS;
  float* __restrict__ orow = out + (size_t)row * NCOLS;

  int k = *kptr;
  if (k < 1) k = 1;
  if (k > NCOLS) k = NCOLS;

  // ---------------- 1) Async DMA row into LDS ----------------
#pragma unroll
  for (int j = 0; j < NCOLS / (TPB * 4); ++j) {
    unsigned idx = (unsigned)(j * TPB + tid) * 4u;   // float4-aligned element idx
    unsigned ldsw = swzg(idx >> 2) << 2;             // swizzled word index
#if HAVE_ASYNC_LDS
    __builtin_amdgcn_global_load_async_to_lds_b128(
        (g_i4p)(xrow + idx), (l_i4p)(rowbuf + ldsw), 0, 0);
#else
    *(float4v*)(rowbuf + ldsw) = *(const float4v*)(xrow + idx);
#endif
  }
  wait_async0();
  __syncthreads();

  // ---------------- 2) Radix select: k-th largest key ----------------
  unsigned prefix = 0;       // selected high digits so far
  unsigned rem = (unsigned)k;

  for (int d = 3; d >= 0; --d) {
    // clear per-wave histograms
    for (int i = tid; i < WAVES * HSTRIDE; i += TPB) whist[i] = 0u;
    __syncthreads();

    const int shift = 8 * d;
    unsigned* myh = &whist[wv * HSTRIDE];
    for (int j = 0; j < NCOLS / TPB; ++j) {
      unsigned idx = (unsigned)(j * TPB) + tid;      // stride-256: conflict-free
      unsigned key = key_of(rowbuf[swzw(idx)]);
      bool match = (d == 3) || ((key >> (shift + 8)) == prefix);
      if (match) atomicAdd(&myh[(key >> shift) & 255u], 1u);  // ds_add_u32
    }
    __syncthreads();

    // reduce the 8 wave histograms
    unsigned s = 0;
#pragma unroll
    for (int w = 0; w < WAVES; ++w) s += whist[w * HSTRIDE + tid];
    hist[tid] = s;
    __syncthreads();

    // inclusive suffix scan: hist[v] := count of candidates with digit >= v
    for (int off = 1; off < TPB; off <<= 1) {
      unsigned v = hist[tid];
      unsigned a = (tid + off < TPB) ? hist[tid + off] : 0u;
      __syncthreads();
      hist[tid] = v + a;
      __syncthreads();
    }
    unsigned sufv = hist[tid];
    unsigned sufn = (tid < TPB - 1) ? hist[tid + 1] : 0u;
    if (sufv >= rem && sufn < rem) sel = tid;  // exactly one thread matches
    __syncthreads();

    unsigned v = sel;
    unsigned greater = (v < 255u) ? hist[v + 1] : 0u;  // broadcast read
    rem -= greater;
    prefix = (prefix << 8) | v;
    __syncthreads();
  }

  const unsigned T = prefix;      // key of the k-th largest element
  const unsigned need_eq = rem;   // how many elements equal to T to keep

  // ---------------- 3) Index-ordered tie resolution ----------------
  // thread t owns contiguous chunk [t*32, t*32+32) so ranks follow index order
  // (matches jax.lax.top_k's first-occurrence tie behavior).
  const unsigned base = tid * (NCOLS / TPB);
  unsigned c = 0;
  for (int j = 0; j < NCOLS / TPB; ++j)
    c += (key_of(rowbuf[swzw(base + (unsigned)j)]) == T) ? 1u : 0u;
  eqcnt[tid] = c;
  __syncthreads();
  for (int off = 1; off < TPB; off <<= 1) {  // inclusive forward scan
    unsigned v = eqcnt[tid];
    unsigned a = (tid >= (unsigned)off) ? eqcnt[tid - off] : 0u;
    __syncthreads();
    eqcnt[tid] = v + a;
    __syncthreads();
  }
  unsigned rank = eqcnt[tid] - c;  // exclusive prefix of equals before my chunk

  // ---------------- 4) Resolve mask in-place in LDS ----------------
  for (int j = 0; j < NCOLS / TPB; ++j) {
    unsigned w = swzw(base + (unsigned)j);
    float xv = rowbuf[w];
    float val = xv > 0.0f ? xv : 0.0f;   // relu
    unsigned key = key_of(xv);
    float o = 0.0f;
    if (key > T) {
      o = val;
    } else if (key == T) {
      if (rank < need_eq) o = val;
      ++rank;
    }
    rowbuf[w] = o;
  }
  __syncthreads();

  // ---------------- 5) Async DMA row LDS -> HBM ----------------
#pragma unroll
  for (int j = 0; j < NCOLS / (TPB * 4); ++j) {
    unsigned idx = (unsigned)(j * TPB + tid) * 4u;
    unsigned ldsw = swzg(idx >> 2) << 2;
#if HAVE_ASYNC_LDS
    __builtin_amdgcn_global_store_async_from_lds_b128(
        (g_i4p)(orow + idx), (l_i4p)(rowbuf + ldsw), 0, 0);
#else
    *(float4v*)(orow + idx) = *(const float4v*)(rowbuf + ldsw);
#endif
  }
  wait_async0();  // S_ENDPGM also does implicit wait-idle
}

extern "C" void kernel_launch(void* const* d_in, const int* in_sizes, int n_in,
                              void* d_out, int out_size, void* d_ws, size_t ws_size,
                              hipStream_t stream) {
  const float* x = (const float*)d_in[0];
  const int* kptr = (const int*)d_in[1];
  float* out = (float*)d_out;
  int rows = in_sizes[0] / NCOLS;
  topk_mask_kernel<<<dim3((unsigned)rows), dim3(TPB), 0, stream>>>(x, kptr, out);
}